// LCAEdgeLoss_48524540510500
// MI455X (gfx1250) — compile-verified
//
#include <hip/hip_runtime.h>
#include <hip/hip_bf16.h>

typedef float v2f __attribute__((ext_vector_type(2)));
typedef float v8f __attribute__((ext_vector_type(8)));

#define BB 16384
#define CC 1000
#define ROWS 32                 // rows per block = one wave32
#define STRIDE 33               // padded column stride (words) -> 2-way conflicts on scatter
#define SLAB_WORDS (CC * STRIDE)        // 33000
#define SLAB_PAD   33024                // rounded up to multiple of 64 for WMMA loop
#define PSLAB_OFF  SLAB_PAD
#define SMEM_WORDS (SLAB_PAD + 1024)    // + parent table
// dynamic LDS: 136,192 bytes -> 2 blocks per 320KB WGP

__global__ __launch_bounds__(32) void lca_main(const float* __restrict__ outputs,
                                               const float* __restrict__ targets,
                                               const int*   __restrict__ parent,
                                               float*       __restrict__ partials)
{
    extern __shared__ unsigned smem[];
    unsigned* Lbits = smem;
    int*      pslab = (int*)(smem + PSLAB_OFF);

    const int t  = threadIdx.x;          // 0..31 (lane == row-in-tile)
    const int b0 = blockIdx.x * ROWS;

    // phase 0: zero pad words, stage parent[] into LDS
    for (int j = t; j < CC; j += 32) {
        Lbits[j * STRIDE + 32] = 0u;     // per-column pad word
        pslab[j] = parent[j];
    }
    if (t < SLAB_PAD - SLAB_WORDS) Lbits[SLAB_WORDS + t] = 0u;  // tail pad

    // phase 1: coalesced float4 streaming loads, softplus, transposed scatter to LDS.
    // Tile (32 rows x 1000 cols) is contiguous in global memory; 16B aligned (4000B row).
    const float4* o4 = reinterpret_cast<const float4*>(outputs + (size_t)b0 * CC);
    const float4* t4 = reinterpret_cast<const float4*>(targets + (size_t)b0 * CC);
    for (int i = 0; i < (ROWS * CC) / 128; ++i) {   // 250 iterations
        float4 xo = o4[i * 32 + t];
        float4 xt = t4[i * 32 + t];
        const int flat0 = i * 128 + t * 4;          // a float4 never crosses a row (1000 % 4 == 0)
        #pragma unroll
        for (int e = 0; e < 4; ++e) {
            float x  = (&xo.x)[e];
            float tg = (&xt.x)[e];
            int flat = flat0 + e;
            int r = flat / CC;                      // row in tile
            int j = flat - r * CC;                  // column
            // loss0 = logaddexp(0,x) - t*x, numerically stable softplus
            float sp = fmaxf(x, 0.0f) + log1pf(expf(-fabsf(x)));
            float v  = sp - tg * x;
            // steal mantissa LSB to carry the (target==1) flag: avoids a 32KB flag slab
            unsigned u = (__float_as_uint(v) & ~1u) | (tg != 0.0f ? 1u : 0u);
            Lbits[j * STRIDE + r] = u;
        }
    }
    __syncthreads();

    // phase 2: sequential tree scan over columns; rows parallel across lanes.
    // parent[j] < j guaranteed, so in-order walk uses finalized parent values.
    for (int j = 1; j < CC; ++j) {
        int p = pslab[j];
        if (p < 0) continue;
        unsigned uj = Lbits[j * STRIDE + t];
        float vj = __uint_as_float(uj);
        float vp = __uint_as_float(Lbits[p * STRIDE + t]);
        float nv = (uj & 1u) ? fmaxf(vj, vp) : (vj + vp);   // t==1 -> max, t==0 -> add
        Lbits[j * STRIDE + t] = __float_as_uint(nv);
    }
    __syncthreads();

    // phase 3: slab reduction on the matrix pipe.
    // A = ones(16x4)  =>  D[m][n] = sum_k B[k][n] + C[m][n]; all 16 rows identical,
    // so sum(all D) == 16 * sum(slab chunk). Pads are +0.0f so they contribute nothing.
    const float* Lf = reinterpret_cast<const float*>(Lbits);
    v2f a; a[0] = 1.0f; a[1] = 1.0f;
    v8f acc = {0.f, 0.f, 0.f, 0.f, 0.f, 0.f, 0.f, 0.f};
    for (int base = 0; base < SLAB_PAD; base += 64) {       // 516 accumulating WMMAs
        v2f b;
        b[0] = Lf[base + t];
        b[1] = Lf[base + 32 + t];
        acc = __builtin_amdgcn_wmma_f32_16x16x4_f32(
                  /*neg_a=*/false, a, /*neg_b=*/false, b,
                  /*c_mod=*/(short)0, acc, /*reuse_a=*/false, /*reuse_b=*/false);
    }
    float s = acc[0] + acc[1] + acc[2] + acc[3] + acc[4] + acc[5] + acc[6] + acc[7];
    #pragma unroll
    for (int off = 16; off > 0; off >>= 1)
        s += __shfl_xor(s, off, 32);
    if (t == 0) partials[blockIdx.x] = s * (1.0f / 16.0f);
}

__global__ __launch_bounds__(32) void lca_finish(const float* __restrict__ partials,
                                                 float* __restrict__ out)
{
    const int t = threadIdx.x;
    float s = 0.0f;
    for (int k = t; k < BB / ROWS; k += 32) s += partials[k];   // fixed order: deterministic
    #pragma unroll
    for (int off = 16; off > 0; off >>= 1)
        s += __shfl_xor(s, off, 32);
    if (t == 0) out[0] = s * (1.0f / ((float)BB * (float)CC));
}

extern "C" void kernel_launch(void* const* d_in, const int* in_sizes, int n_in,
                              void* d_out, int out_size, void* d_ws, size_t ws_size,
                              hipStream_t stream)
{
    (void)in_sizes; (void)n_in; (void)out_size; (void)ws_size;
    const float* outputs = (const float*)d_in[0];   // (16384, 1000) f32
    const float* targets = (const float*)d_in[1];   // (16384, 1000) f32
    const int*   parent  = (const int*)d_in[2];     // (1000,) i32
    float*       partials = (float*)d_ws;           // 512 floats, fully overwritten each call

    const size_t smem_bytes = SMEM_WORDS * sizeof(unsigned);    // 136,192 B dynamic LDS
    lca_main<<<dim3(BB / ROWS), dim3(32), smem_bytes, stream>>>(outputs, targets, parent, partials);
    lca_finish<<<dim3(1), dim3(32), 0, stream>>>(partials, (float*)d_out);
}